// PLM_GINConvNet_2439541424582
// MI455X (gfx1250) — compile-verified
//
#include <hip/hip_runtime.h>

#define NNODES   100000
#define NEDGES   3200000
#define NGRAPHS  2048
#define GDIM     32
#define FIN      78
#define FINP     80        // FIN padded to multiple of 4
#define EMBED    320
#define OUTD     128
#define PLMOUT   128
#define BN_EPS   1e-5f

typedef __attribute__((ext_vector_type(2))) float v2f;
typedef __attribute__((ext_vector_type(8))) float v8f;

// ---------------- elementwise helpers ----------------

__global__ void zero_kernel(float* __restrict__ p, int n) {
    for (int i = blockIdx.x * blockDim.x + threadIdx.x; i < n; i += gridDim.x * blockDim.x)
        p[i] = 0.0f;
}

// dst[0:n) = src[0:n), dst[n:ntotal) = 0   (flat zero-pad copy, used for w1_0)
__global__ void copy_pad_flat_kernel(float* __restrict__ dst, const float* __restrict__ src,
                                     int n, int ntotal) {
    for (int i = blockIdx.x * blockDim.x + threadIdx.x; i < ntotal; i += gridDim.x * blockDim.x)
        dst[i] = (i < n) ? src[i] : 0.0f;
}

// dst (M x Kp, row-major) = src (M x K) with zero-padded tail columns
__global__ void copy_pad_rows_kernel(float* __restrict__ dst, const float* __restrict__ src,
                                     int M, int K, int Kp) {
    int total = M * Kp;
    for (int i = blockIdx.x * blockDim.x + threadIdx.x; i < total; i += gridDim.x * blockDim.x) {
        int n = i / Kp;
        int f = i - n * Kp;
        dst[i] = (f < K) ? src[(size_t)n * K + f] : 0.0f;
    }
}

// ---------------- GIN edge aggregation: h[dst] += x[src] ----------------
// x has row stride xs_stride (=din), h has row stride h_stride (=din padded)

__global__ void edge_scatter_kernel(const float* __restrict__ x,
                                    const int* __restrict__ srci,
                                    const int* __restrict__ dsti,
                                    float* __restrict__ h,
                                    int din, int xs_stride, int h_stride) {
    for (int e = blockIdx.x * blockDim.x + threadIdx.x; e < NEDGES; e += gridDim.x * blockDim.x) {
        int s = srci[e];
        int d = dsti[e];
        const float* xs = x + (size_t)s * xs_stride;
        float* hd = h + (size_t)d * h_stride;
        for (int f = 0; f < din; ++f)
            atomicAdd(&hd[f], xs[f]);
    }
}

// ---------------- global_add_pool: pooled[batch[n]] += x[n] ----------------

__global__ void pool_kernel(const float* __restrict__ x,
                            const int* __restrict__ batch,
                            float* __restrict__ pooled) {
    for (int n = blockIdx.x * blockDim.x + threadIdx.x; n < NNODES; n += gridDim.x * blockDim.x) {
        int g = batch[n];
        const float* xn = x + (size_t)n * GDIM;
        float* pg = pooled + (size_t)g * GDIM;
        for (int f = 0; f < GDIM; ++f)
            atomicAdd(&pg[f], xn[f]);
    }
}

// ---------------- WMMA f32 GEMM: C = [relu](A(Mx K4) @ W(K4 x N) + bias) ----------------
// Requirements (guaranteed by caller): M multiple of 16, N multiple of 16,
// K4 multiple of 4, A valid for lda >= K4 with zero pad, W valid for K4 rows.
// One wave per 16x16 output tile via V_WMMA_F32_16X16X4_F32.
// A 16x4 layout: lane m = lane%16; lanes 0-15 hold K={k,k+1}, lanes 16-31 K={k+2,k+3}.
// B 4x16 layout: lane n = lane%16; same K split.
// C/D 16x16:     VGPR v -> row = (lane<16 ? v : v+8), col = lane%16.

__global__ __launch_bounds__(32) void gemm_tile_kernel(
    const float* __restrict__ A, const float* __restrict__ W,
    const float* __restrict__ bias, float* __restrict__ C,
    int lda, int K4, int N, int ldc, int do_relu) {

    int tilesN = N >> 4;
    int tm = blockIdx.x / tilesN;
    int tn = blockIdx.x - tm * tilesN;

    int lane = threadIdx.x;
    int half = lane >> 4;      // 0: K pair {0,1}; 1: K pair {2,3}
    int l16  = lane & 15;
    int col  = tn * 16 + l16;

    const float* Arow = A + (size_t)(tm * 16 + l16) * lda + half * 2;  // 8B aligned
    const float* Wcol = W + (size_t)(half * 2) * N + col;

    v8f acc = {0.f, 0.f, 0.f, 0.f, 0.f, 0.f, 0.f, 0.f};

#pragma unroll 4
    for (int k0 = 0; k0 < K4; k0 += 4) {
        v2f a = *(const v2f*)(Arow + k0);          // global_load_b64
        v2f b;
        b.x = Wcol[(size_t)k0 * N];
        b.y = Wcol[(size_t)(k0 + 1) * N];
        acc = __builtin_amdgcn_wmma_f32_16x16x4_f32(
            /*neg_a=*/false, a, /*neg_b=*/false, b,
            /*c_mod=*/(short)0, acc, /*reuse_a=*/false, /*reuse_b=*/false);
    }

    float bv = bias[col];
    float* Cp = C + (size_t)(tm * 16 + half * 8) * ldc + col;
#pragma unroll
    for (int v = 0; v < 8; ++v) {
        float val = acc[v] + bv;
        if (do_relu) val = fmaxf(val, 0.f);
        Cp[(size_t)v * ldc] = val;
    }
}

// ---------------- BatchNorm (training-mode batch stats) ----------------
// sums[0:F] = sum, sums[F:2F] = sum of squares; blockDim must be multiple of F.

__global__ void bn_stats_kernel(const float* __restrict__ h, int M, int F,
                                float* __restrict__ sums) {
    extern __shared__ float sm[];             // 2 * blockDim floats
    int t = threadIdx.x;
    int f = t % F;
    int g = t / F;
    int groups = blockDim.x / F;
    float s = 0.f, s2 = 0.f;
    for (int n = blockIdx.x * groups + g; n < M; n += gridDim.x * groups) {
        float v = h[(size_t)n * F + f];
        s += v; s2 += v * v;
    }
    sm[t] = s;
    sm[blockDim.x + t] = s2;
    __syncthreads();
    if (g == 0) {
        for (int gg = 1; gg < groups; ++gg) {
            s  += sm[gg * F + f];
            s2 += sm[blockDim.x + gg * F + f];
        }
        atomicAdd(&sums[f], s);
        atomicAdd(&sums[F + f], s2);
    }
}

__global__ void bn_apply_kernel(const float* __restrict__ h, const float* __restrict__ sums,
                                const float* __restrict__ gamma, const float* __restrict__ beta,
                                float* __restrict__ out, int M, int F, int ldo,
                                float invM, int do_relu) {
    int total = M * F;
    for (int i = blockIdx.x * blockDim.x + threadIdx.x; i < total; i += gridDim.x * blockDim.x) {
        int n = i / F;
        int f = i - n * F;
        float mean = sums[f] * invM;
        float var  = sums[F + f] * invM - mean * mean;  // biased var (matches jnp.var)
        float v = gamma[f] * (h[i] - mean) * rsqrtf(var + BN_EPS) + beta[f];
        if (do_relu) v = fmaxf(v, 0.f);
        out[(size_t)n * ldo + f] = v;
    }
}

// ---------------- final 256 -> 1 projection ----------------

__global__ void out_kernel(const float* __restrict__ f2, const float* __restrict__ w,
                           const float* __restrict__ b, float* __restrict__ out) {
    int g = blockIdx.x * blockDim.x + threadIdx.x;
    if (g < NGRAPHS) {
        float s = b[0];
        const float* r = f2 + (size_t)g * 256;
        for (int k = 0; k < 256; ++k) s += r[k] * w[k];
        out[g] = s;
    }
}

// ---------------- driver ----------------

extern "C" void kernel_launch(void* const* d_in, const int* in_sizes, int n_in,
                              void* d_out, int out_size, void* d_ws, size_t ws_size,
                              hipStream_t stream) {
    const float* x      = (const float*)d_in[0];
    const int*   ei     = (const int*)  d_in[1];
    const int*   batch  = (const int*)  d_in[2];
    const float* target = (const float*)d_in[3];

    // params dict flattened in insertion order:
    // per layer l (0..4): w1,b1,w2,b2,gamma,beta  -> P[6l..6l+5]
    // then wxd,bxd,wxt,bxt,gamma_xt,beta_xt,w_fc1,b_fc1,w_fc2,b_fc2,w_out,b_out
    const float* P[42];
    for (int i = 0; i < 42; ++i) P[i] = (const float*)d_in[4 + i];
    const float* wxd = P[30]; const float* bxd  = P[31];
    const float* wxt = P[32]; const float* bxt  = P[33];
    const float* gxt = P[34]; const float* bexx = P[35];
    const float* wf1 = P[36]; const float* bf1  = P[37];
    const float* wf2 = P[38]; const float* bf2  = P[39];
    const float* wo  = P[40]; const float* bo   = P[41];

    // workspace layout (floats)
    float* ws     = (float*)d_ws;
    float* h      = ws;                                   // NNODES * FINP (max padded din)
    float* h1     = h    + (size_t)NNODES * FINP;         // NNODES * GDIM
    float* xcur   = h1   + (size_t)NNODES * GDIM;         // NNODES * GDIM
    float* w1pad  = xcur + (size_t)NNODES * GDIM;         // FINP * GDIM
    float* sums   = w1pad + (size_t)FINP * GDIM;          // 2*PLMOUT max
    float* pooled = sums + 2 * PLMOUT;                    // NGRAPHS * GDIM
    float* xt_t   = pooled + (size_t)NGRAPHS * GDIM;      // NGRAPHS * PLMOUT
    float* xc     = xt_t + (size_t)NGRAPHS * PLMOUT;      // NGRAPHS * 256
    float* f1     = xc   + (size_t)NGRAPHS * 256;         // NGRAPHS * 1024
    float* f2     = f1   + (size_t)NGRAPHS * 1024;        // NGRAPHS * 256

    const int* srci = ei;
    const int* dsti = ei + NEDGES;

    // zero-pad w1_0 (78x32) -> (80x32) once per launch (deterministic)
    copy_pad_flat_kernel<<<(FINP * GDIM + 255) / 256, 256, 0, stream>>>(
        w1pad, P[0], FIN * GDIM, FINP * GDIM);

    const int tilesM = NNODES / 16;   // 6250, exact

    for (int l = 0; l < 5; ++l) {
        int din  = (l == 0) ? FIN  : GDIM;
        int dinp = (l == 0) ? FINP : GDIM;
        const float* Xin = (l == 0) ? x : xcur;
        const float* w1 = (l == 0) ? w1pad : P[6 * l + 0];
        const float* b1 = P[6 * l + 1];
        const float* w2 = P[6 * l + 2]; const float* b2 = P[6 * l + 3];
        const float* ga = P[6 * l + 4]; const float* be = P[6 * l + 5];

        // h = (1+eps)*x + sum_{j->i} x_j   (eps = 0), with zero-padded columns
        copy_pad_rows_kernel<<<2048, 256, 0, stream>>>(h, Xin, NNODES, din, dinp);
        edge_scatter_kernel<<<4096, 256, 0, stream>>>(Xin, srci, dsti, h, din, din, dinp);

        // h1 = relu(h @ w1 + b1); h = relu(h1 @ w2 + b2)
        gemm_tile_kernel<<<tilesM * (GDIM / 16), 32, 0, stream>>>(
            h, w1, b1, h1, dinp, dinp, GDIM, GDIM, 1);
        gemm_tile_kernel<<<tilesM * (GDIM / 16), 32, 0, stream>>>(
            h1, w2, b2, h, GDIM, GDIM, GDIM, GDIM, 1);

        // xcur = batchnorm(h)
        zero_kernel<<<1, 256, 0, stream>>>(sums, 2 * GDIM);
        bn_stats_kernel<<<256, 256, 2 * 256 * sizeof(float), stream>>>(h, NNODES, GDIM, sums);
        bn_apply_kernel<<<2048, 256, 0, stream>>>(h, sums, ga, be, xcur,
                                                  NNODES, GDIM, GDIM, 1.0f / NNODES, 0);
    }

    // pooled = global_add_pool(xcur, batch)
    zero_kernel<<<64, 256, 0, stream>>>(pooled, NGRAPHS * GDIM);
    pool_kernel<<<1024, 256, 0, stream>>>(xcur, batch, pooled);

    // xd = relu(pooled @ wxd + bxd) -> xc[:, 0:128]
    gemm_tile_kernel<<<(NGRAPHS / 16) * (OUTD / 16), 32, 0, stream>>>(
        pooled, wxd, bxd, xc, GDIM, GDIM, OUTD, 256, 1);

    // xt = relu(BN(target @ wxt + bxt)) -> xc[:, 128:256]
    gemm_tile_kernel<<<(NGRAPHS / 16) * (PLMOUT / 16), 32, 0, stream>>>(
        target, wxt, bxt, xt_t, EMBED, EMBED, PLMOUT, PLMOUT, 0);
    zero_kernel<<<1, 256, 0, stream>>>(sums, 2 * PLMOUT);
    bn_stats_kernel<<<128, 256, 2 * 256 * sizeof(float), stream>>>(xt_t, NGRAPHS, PLMOUT, sums);
    bn_apply_kernel<<<512, 256, 0, stream>>>(xt_t, sums, gxt, bexx, xc + OUTD,
                                             NGRAPHS, PLMOUT, 256, 1.0f / NGRAPHS, 1);

    // fc1 / fc2 / out
    gemm_tile_kernel<<<(NGRAPHS / 16) * (1024 / 16), 32, 0, stream>>>(
        xc, wf1, bf1, f1, 256, 256, 1024, 1024, 1);
    gemm_tile_kernel<<<(NGRAPHS / 16) * (256 / 16), 32, 0, stream>>>(
        f1, wf2, bf2, f2, 1024, 1024, 256, 256, 1);
    out_kernel<<<(NGRAPHS + 255) / 256, 256, 0, stream>>>(f2, wo, bo, (float*)d_out);
}